// VSiFuModel_46127948759549
// MI455X (gfx1250) — compile-verified
//
#include <hip/hip_runtime.h>
#include <hip/hip_bf16.h>
#include <math.h>

// Problem constants (from reference)
#define Bn 16
#define Sn 32
#define Tn 32
#define Dn 128
#define Ln 64
#define STEPSn 128
#define Nn 4096
#define SK_ITERS_C 100
#define NS_ITERS 30

typedef float v2f __attribute__((ext_vector_type(2)));
typedef float v8f __attribute__((ext_vector_type(8)));

// ---------------------------------------------------------------------------
// f32 WMMA 16x16 tile MAC:  D = A(16x4) * B(4x16) + C   (wave32, CDNA5)
// ---------------------------------------------------------------------------
__device__ __forceinline__ v8f wmma4(v2f a, v2f b, v8f c) {
  return __builtin_amdgcn_wmma_f32_16x16x4_f32(
      /*neg_a=*/false, a, /*neg_b=*/false, b,
      /*c_mod=*/(short)0, c, /*reuse_a=*/false, /*reuse_b=*/false);
}

// ===========================================================================
// 1) gi = history @ W_ih^T + b_ih     (2048 x 384, K=128)  [WMMA]
// ===========================================================================
__global__ __launch_bounds__(32) void k_gemm_gi(
    const float* __restrict__ hist, const float* __restrict__ Wih,
    const float* __restrict__ bih, float* __restrict__ gi) {
  const int lane = threadIdx.x & 31;
  const int l16 = lane & 15, hi = lane >> 4;
  const int mrow = blockIdx.x * 16;   // over B*STEPS = 2048
  const int ncol = blockIdx.y * 16;   // over 3D = 384
  const int row = mrow + l16, col = ncol + l16;
  v8f acc = {};
  for (int k = 0; k < Dn; k += 4) {
    const int kk = k + (hi ? 2 : 0);
    v2f a, b;
    a.x = hist[(size_t)row * Dn + kk];
    a.y = hist[(size_t)row * Dn + kk + 1];
    b.x = Wih[(size_t)col * Dn + kk];      // B[k][n] = W_ih[n][k]
    b.y = Wih[(size_t)col * Dn + kk + 1];
    acc = wmma4(a, b, acc);
  }
  for (int r = 0; r < 8; ++r) {
    const int orow = mrow + r + (hi ? 8 : 0);
    gi[(size_t)orow * 384 + col] = acc[r] + bih[col];
  }
}

// ===========================================================================
// 2) GRU scan (sequential, per-batch block of 128 threads)
// ===========================================================================
__global__ __launch_bounds__(128) void k_gru_scan(
    const float* __restrict__ gi, const float* __restrict__ Whh,
    const float* __restrict__ bhh, float* __restrict__ summary) {
  __shared__ float h[Dn];
  const int b = blockIdx.x, j = threadIdx.x;
  h[j] = 0.f;
  __syncthreads();
  for (int t = 0; t < STEPSn; ++t) {
    float a0 = bhh[j], a1 = bhh[j + Dn], a2 = bhh[j + 2 * Dn];
    for (int k = 0; k < Dn; ++k) {
      const float hk = h[k];
      a0 += hk * Whh[(size_t)j * Dn + k];
      a1 += hk * Whh[(size_t)(j + Dn) * Dn + k];
      a2 += hk * Whh[(size_t)(j + 2 * Dn) * Dn + k];
    }
    const float* g = gi + ((size_t)b * STEPSn + t) * 384;
    const float r = 1.f / (1.f + expf(-(g[j] + a0)));
    const float z = 1.f / (1.f + expf(-(g[j + Dn] + a1)));
    const float n = tanhf(g[j + 2 * Dn] + r * a2);
    const float hn = (1.f - z) * n + z * h[j];
    __syncthreads();
    h[j] = hn;
    __syncthreads();
  }
  summary[(size_t)b * Dn + j] = h[j];
}

// ===========================================================================
// 3) latent: z = mu + eps*exp(0.5*logvar)
// ===========================================================================
__global__ __launch_bounds__(Ln) void k_latent(
    const float* __restrict__ summary, const float* __restrict__ Wmu,
    const float* __restrict__ bmu, const float* __restrict__ Wlv,
    const float* __restrict__ blv, const float* __restrict__ eps,
    float* __restrict__ zws, float* __restrict__ out_z) {
  const int b = blockIdx.x, l = threadIdx.x;
  const float* s = summary + (size_t)b * Dn;
  float mu = bmu[l], lv = blv[l];
  for (int k = 0; k < Dn; ++k) {
    mu += s[k] * Wmu[(size_t)l * Dn + k];
    lv += s[k] * Wlv[(size_t)l * Dn + k];
  }
  const float zz = mu + eps[(size_t)b * Ln + l] * expf(0.5f * lv);
  zws[(size_t)b * Ln + l] = zz;
  out_z[(size_t)b * Ln + l] = zz;
}

// psi_z[b][d] = sum_l W_psi[l][d]*z[l] ; u_z[b][d] = sum_l W_costU[d][l]*z[l]
__global__ __launch_bounds__(Dn) void k_zproj(
    const float* __restrict__ z, const float* __restrict__ Wpsi,
    const float* __restrict__ WcU, float* __restrict__ psiz,
    float* __restrict__ uz) {
  const int b = blockIdx.x, d = threadIdx.x;
  const float* zb = z + (size_t)b * Ln;
  float p = 0.f, u = 0.f;
  for (int l = 0; l < Ln; ++l) {
    p += Wpsi[(size_t)l * Dn + d] * zb[l];
    u += WcU[(size_t)d * Ln + l] * zb[l];
  }
  psiz[(size_t)b * Dn + d] = p;
  uz[(size_t)b * Dn + d] = u;
}

// ===========================================================================
// 4) source / target distributions
// ===========================================================================
__global__ __launch_bounds__(Sn) void k_src_dist(
    const float* __restrict__ sig, float* __restrict__ out_src) {
  __shared__ float e[Sn];
  const int b = blockIdx.x, s = threadIdx.x;
  const float* x = sig + ((size_t)b * Sn + s) * Dn;
  float acc = 0.f;
  for (int d = 0; d < Dn; ++d) acc += x[d] * x[d];
  e[s] = sqrtf(acc);  // TEMP = 1.0
  __syncthreads();
  float m = -INFINITY;
  for (int i = 0; i < Sn; ++i) m = fmaxf(m, e[i]);
  float sum = 0.f;
  for (int i = 0; i < Sn; ++i) sum += expf(e[i] - m);
  out_src[(size_t)b * Sn + s] = expf(e[s] - m) / sum;
}

__global__ __launch_bounds__(Tn) void k_tgt_dist(
    const int* __restrict__ cand, const float* __restrict__ protos,
    const float* __restrict__ psiz, float* __restrict__ out_tgt) {
  __shared__ float e[Tn];
  const int b = blockIdx.x, t = threadIdx.x;
  const int idx = cand[(size_t)b * Tn + t];
  const float* p = protos + (size_t)idx * Dn;
  const float* q = psiz + (size_t)b * Dn;
  float acc = 0.f;
  for (int d = 0; d < Dn; ++d) acc += p[d] * q[d];
  e[t] = acc;
  __syncthreads();
  float m = -INFINITY;
  for (int i = 0; i < Tn; ++i) m = fmaxf(m, e[i]);
  float sum = 0.f;
  for (int i = 0; i < Tn; ++i) sum += expf(e[i] - m);
  out_tgt[(size_t)b * Tn + t] = expf(e[t] - m) / sum;
}

// ===========================================================================
// 5) A_s = src_sig @ W_costA^T (512x128,K=128)  [WMMA]
//    Bp  = protos[cand] @ W_costB^T + U_z       [WMMA + gather]
// ===========================================================================
__global__ __launch_bounds__(32) void k_gemm_As(
    const float* __restrict__ sig, const float* __restrict__ WcA,
    float* __restrict__ As) {
  const int lane = threadIdx.x & 31, l16 = lane & 15, hi = lane >> 4;
  const int mrow = blockIdx.x * 16, ncol = blockIdx.y * 16;
  const int row = mrow + l16, col = ncol + l16;
  v8f acc = {};
  for (int k = 0; k < Dn; k += 4) {
    const int kk = k + (hi ? 2 : 0);
    v2f a, b;
    a.x = sig[(size_t)row * Dn + kk];
    a.y = sig[(size_t)row * Dn + kk + 1];
    b.x = WcA[(size_t)col * Dn + kk];
    b.y = WcA[(size_t)col * Dn + kk + 1];
    acc = wmma4(a, b, acc);
  }
  for (int r = 0; r < 8; ++r) {
    const int orow = mrow + r + (hi ? 8 : 0);
    As[(size_t)orow * Dn + col] = acc[r];
  }
}

__global__ __launch_bounds__(32) void k_gemm_Bp(
    const int* __restrict__ cand, const float* __restrict__ protos,
    const float* __restrict__ WcB, const float* __restrict__ uz,
    float* __restrict__ Bp) {
  const int lane = threadIdx.x & 31, l16 = lane & 15, hi = lane >> 4;
  const int mrow = blockIdx.x * 16, ncol = blockIdx.y * 16;
  const int row = mrow + l16, col = ncol + l16;
  const int idx = cand[row];  // per-lane gather of prototype row
  v8f acc = {};
  for (int k = 0; k < Dn; k += 4) {
    const int kk = k + (hi ? 2 : 0);
    v2f a, b;
    a.x = protos[(size_t)idx * Dn + kk];
    a.y = protos[(size_t)idx * Dn + kk + 1];
    b.x = WcB[(size_t)col * Dn + kk];
    b.y = WcB[(size_t)col * Dn + kk + 1];
    acc = wmma4(a, b, acc);
  }
  for (int r = 0; r < 8; ++r) {
    const int orow = mrow + r + (hi ? 8 : 0);
    const int bb = orow >> 5;  // batch of this row
    Bp[(size_t)orow * Dn + col] = acc[r] + uz[(size_t)bb * Dn + col];
  }
}

// squared norms of A_s rows and Bp rows
__global__ __launch_bounds__(64) void k_norms(
    const float* __restrict__ As, const float* __restrict__ Bp,
    float* __restrict__ sa, float* __restrict__ sb) {
  const int b = blockIdx.x, tid = threadIdx.x;
  const int i = tid & 31;
  const float* x = (tid < 32) ? (As + ((size_t)b * Sn + i) * Dn)
                              : (Bp + ((size_t)b * Tn + i) * Dn);
  float acc = 0.f;
  for (int d = 0; d < Dn; ++d) acc += x[d] * x[d];
  if (tid < 32) sa[(size_t)b * Sn + i] = acc;
  else          sb[(size_t)b * Tn + i] = acc;
}

// cost[b,s,t] = sa[s] + sb[t] - 2 * (A_s[b] @ Bp[b]^T)   [WMMA, 32x32 K=128]
__global__ __launch_bounds__(32) void k_gemm_cost(
    const float* __restrict__ As, const float* __restrict__ Bp,
    const float* __restrict__ sa, const float* __restrict__ sb,
    float* __restrict__ cost) {
  const int lane = threadIdx.x & 31, l16 = lane & 15, hi = lane >> 4;
  const int b = blockIdx.x;
  const int mrow = blockIdx.y * 16, ncol = blockIdx.z * 16;
  const int row = mrow + l16, col = ncol + l16;
  const float* Ab = As + (size_t)b * Sn * Dn;
  const float* Bb = Bp + (size_t)b * Tn * Dn;
  v8f acc = {};
  for (int k = 0; k < Dn; k += 4) {
    const int kk = k + (hi ? 2 : 0);
    v2f a, bf;
    a.x = Ab[(size_t)row * Dn + kk];
    a.y = Ab[(size_t)row * Dn + kk + 1];
    bf.x = Bb[(size_t)col * Dn + kk];   // B[k][n] = Bp[n][k]
    bf.y = Bb[(size_t)col * Dn + kk + 1];
    acc = wmma4(a, bf, acc);
  }
  for (int r = 0; r < 8; ++r) {
    const int orow = mrow + r + (hi ? 8 : 0);
    cost[(size_t)b * Sn * Tn + orow * Tn + col] =
        sa[(size_t)b * Sn + orow] + sb[(size_t)b * Tn + col] - 2.f * acc[r];
  }
}

// ===========================================================================
// 6) Sinkhorn (log-domain, freeze-on-converge), one 1024-thread block / batch
// ===========================================================================
__global__ __launch_bounds__(1024) void k_sinkhorn(
    const float* __restrict__ cost, const float* __restrict__ mu,
    const float* __restrict__ nu, float* __restrict__ out_T) {
  __shared__ float lK[Sn * Tn], red[Sn * Tn];
  __shared__ float lu[Sn], lv[Tn], nuu[Sn], nvv[Tn], lmu[Sn], lnu[Tn];
  __shared__ float upd;
  __shared__ int done;
  const int b = blockIdx.x, tid = threadIdx.x;
  const int s = tid >> 5, t = tid & 31;
  lK[tid] = -cost[(size_t)b * Sn * Tn + tid] * 10.f;  // 1/EPS_SK
  if (tid < 32) {
    lu[tid] = 0.f; lv[tid] = 0.f;
    lmu[tid] = logf(fmaxf(mu[(size_t)b * Sn + tid], 1.1754943508e-38f));
    lnu[tid] = logf(fmaxf(nu[(size_t)b * Tn + tid], 1.1754943508e-38f));
  }
  if (tid == 0) done = 0;
  __syncthreads();
  for (int it = 0; it < SK_ITERS_C; ++it) {
    red[tid] = lK[tid] + lv[t];
    __syncthreads();
    if (tid < Sn) {  // lse over t for row tid
      float m = -INFINITY;
      for (int j = 0; j < Tn; ++j) m = fmaxf(m, red[tid * Tn + j]);
      float sm = 0.f;
      for (int j = 0; j < Tn; ++j) sm += expf(red[tid * Tn + j] - m);
      nuu[tid] = lmu[tid] - (m + logf(sm));
    }
    __syncthreads();
    red[tid] = lK[tid] + nuu[s];
    __syncthreads();
    if (tid < Tn) {  // lse over s for col tid
      float m = -INFINITY;
      for (int j = 0; j < Sn; ++j) m = fmaxf(m, red[j * Tn + tid]);
      float sm = 0.f;
      for (int j = 0; j < Sn; ++j) sm += expf(red[j * Tn + tid] - m);
      nvv[tid] = lnu[tid] - (m + logf(sm));
    }
    __syncthreads();
    if (tid == 0) {
      float u = 0.f;
      for (int j = 0; j < Sn; ++j) u = fmaxf(u, fabsf(nuu[j] - lu[j]));
      upd = u;
    }
    __syncthreads();
    const int wasdone = done;  // freeze uses *previous* done (mirrors ref)
    __syncthreads();
    if (tid < 32 && !wasdone) { lu[tid] = nuu[tid]; lv[tid] = nvv[tid]; }
    if (tid == 0 && (isnan(upd) || upd < 1e-6f)) done = 1;
    __syncthreads();
  }
  out_T[(size_t)b * Sn * Tn + tid] = expf(lu[s] + lK[tid] + lv[t]);
}

// ===========================================================================
// 7) intermediate[b,s,:] = sig[b,s,:] @ src_lin[idx]   (per-row matrix, VALU)
// ===========================================================================
__global__ __launch_bounds__(Dn) void k_inter(
    const int* __restrict__ sidx, const float* __restrict__ sig,
    const float* __restrict__ slin, float* __restrict__ inter) {
  const int bs = blockIdx.x, e = threadIdx.x;
  const int idx = sidx[bs];
  const float* lin = slin + (size_t)idx * Dn * Dn;
  const float* x = sig + (size_t)bs * Dn;
  float acc = 0.f;
  for (int d = 0; d < Dn; ++d) acc += x[d] * lin[(size_t)d * Dn + e];
  inter[(size_t)bs * Dn + e] = acc;
}

// ===========================================================================
// 8) edge_linear: per (b,t): E = inter[b] (32x128) @ tgt_lin[idx]^T  [WMMA]
//    One wave owns a 16x128 row strip: A fragment loaded once per K-step and
//    reused by 8 WMMAs (8 accumulators = 64 VGPRs).
// ===========================================================================
__global__ __launch_bounds__(32) void k_gemm_edge(
    const int* __restrict__ cand, const float* __restrict__ inter,
    const float* __restrict__ tlin, float* __restrict__ basis) {
  const int lane = threadIdx.x & 31, l16 = lane & 15, hi = lane >> 4;
  const int bt = blockIdx.x, b = bt >> 5, t = bt & 31;
  const int mrow = blockIdx.y * 16;
  const int row = mrow + l16;
  const int idx = cand[bt];
  const float* A = inter + (size_t)b * Sn * Dn;
  const float* Bm = tlin + (size_t)idx * Dn * Dn;  // tgt_T[k][n] = tlin[n][k]
  __builtin_prefetch(Bm + (size_t)l16 * Dn, 0, 1);        // global_prefetch_b8
  __builtin_prefetch(Bm + (size_t)(64 + l16) * Dn, 0, 1);
  v8f acc[8] = {};
  for (int k = 0; k < Dn; k += 4) {
    const int kk = k + (hi ? 2 : 0);
    v2f a;
    a.x = A[(size_t)row * Dn + kk];
    a.y = A[(size_t)row * Dn + kk + 1];
#pragma unroll
    for (int nt = 0; nt < 8; ++nt) {
      const int col = nt * 16 + l16;
      v2f bf;
      bf.x = Bm[(size_t)col * Dn + kk];
      bf.y = Bm[(size_t)col * Dn + kk + 1];
      acc[nt] = wmma4(a, bf, acc[nt]);
    }
  }
#pragma unroll
  for (int nt = 0; nt < 8; ++nt) {
    const int col = nt * 16 + l16;
    for (int r = 0; r < 8; ++r) {
      const int orow = mrow + r + (hi ? 8 : 0);  // s index
      basis[((size_t)(b * Sn + orow) * Tn + t) * Dn + col] = acc[nt][r];
    }
  }
}

// ===========================================================================
// 9) tilde[b,:] = sum_{s,t} P[b,s,t] * (E[b,s,t,:] + src_bias + tgt_bias)
// ===========================================================================
__global__ __launch_bounds__(Dn) void k_tilde(
    const int* __restrict__ sidx, const int* __restrict__ cand,
    const float* __restrict__ sbias, const float* __restrict__ tbias,
    const float* __restrict__ trans, const float* __restrict__ basis,
    float* __restrict__ tilde) {
  const int b = blockIdx.x, d = threadIdx.x;
  float acc = 0.f;
  for (int s = 0; s < Sn; ++s) {
    const float sb_ = sbias[(size_t)sidx[b * Sn + s] * Dn + d];
    for (int t = 0; t < Tn; ++t) {
      const float w = trans[(size_t)b * Sn * Tn + s * Tn + t];
      const float tb_ = tbias[(size_t)cand[b * Tn + t] * Dn + d];
      acc += w * (basis[((size_t)(b * Sn + s) * Tn + t) * Dn + d] + sb_ + tb_);
    }
  }
  tilde[(size_t)b * Dn + d] = acc;
}

// ===========================================================================
// 10) M = basis^T @ basis per batch (128x128, K=1024)  [WMMA]
//     Note: B^T(BB^T)^+B == M M^+ with M = B^T B  (same orthogonal projector),
//     so the reference's 1024x1024 pinv collapses to a 128x128 pseudo-inverse.
// ===========================================================================
__global__ __launch_bounds__(32) void k_gemm_M(
    const float* __restrict__ basis, float* __restrict__ Mm) {
  const int lane = threadIdx.x & 31, l16 = lane & 15, hi = lane >> 4;
  const int b = blockIdx.x;
  const int mrow = blockIdx.y * 16, ncol = blockIdx.z * 16;
  const int row = mrow + l16, col = ncol + l16;
  const float* Bb = basis + (size_t)b * (Sn * Tn) * Dn;
  v8f acc = {};
  for (int k = 0; k < Sn * Tn; k += 4) {
    const int kk = k + (hi ? 2 : 0);
    v2f a, bf;
    a.x = Bb[(size_t)kk * Dn + row];        // A = basis^T
    a.y = Bb[(size_t)(kk + 1) * Dn + row];
    bf.x = Bb[(size_t)kk * Dn + col];
    bf.y = Bb[(size_t)(kk + 1) * Dn + col];
    acc = wmma4(a, bf, acc);
  }
  for (int r = 0; r < 8; ++r) {
    const int orow = mrow + r + (hi ? 8 : 0);
    Mm[((size_t)b * Dn + orow) * Dn + col] = acc[r];
  }
}

// Newton-Schulz init: X0 = M / (||M||_1 * ||M||_inf)  (M symmetric PSD)
__global__ __launch_bounds__(Dn) void k_ns_init(
    const float* __restrict__ Mm, float* __restrict__ X) {
  __shared__ float rs[Dn];
  const int b = blockIdx.x, m = threadIdx.x;
  const float* Mb = Mm + (size_t)b * Dn * Dn;
  float s = 0.f;
  for (int k = 0; k < Dn; ++k) s += fabsf(Mb[(size_t)m * Dn + k]);
  rs[m] = s;
  __syncthreads();
  float mx = 0.f;
  for (int i = 0; i < Dn; ++i) mx = fmaxf(mx, rs[i]);
  const float alpha = (mx > 0.f) ? 1.f / (mx * mx) : 0.f;
  for (int k = 0; k < Dn; ++k)
    X[((size_t)b * Dn + m) * Dn + k] = alpha * Mb[(size_t)m * Dn + k];
}

// ===========================================================================
// 11) Fused Newton-Schulz iteration:  Xn = 2*X - X @ (M @ X)
//     One workgroup (8 waves) per batch. X staged global->LDS with
//     GLOBAL_LOAD_ASYNC_TO_LDS_B128 (+ s_wait_asynccnt); Y lives only in LDS.
//     Stage 2 feeds both WMMA operands from LDS (ds_load path).
// ===========================================================================
__global__ __launch_bounds__(256) void k_ns_iter(
    const float* __restrict__ Mm, const float* __restrict__ X,
    float* __restrict__ Xn) {
  extern __shared__ float smem[];        // 2 * 128*128 floats = 128 KB dynamic
  float* Xs = smem;                      // 128x128
  float* Ys = smem + Dn * Dn;            // 128x128
  const int b = blockIdx.x, tid = threadIdx.x;
  const int wave = tid >> 5, lane = tid & 31;
  const int l16 = lane & 15, hi = lane >> 4;
  const float* Mb = Mm + (size_t)b * Dn * Dn;
  const float* Xb = X + (size_t)b * Dn * Dn;

  // ---- async copy X tile (64 KB) from global into LDS ----
  // Flat aperture rule (ISA 10.2): low 32 bits of a generic LDS pointer are
  // the LDS byte address, so truncation gives the VDST operand.
#pragma unroll
  for (int i = 0; i < 16; ++i) {
    const int elt = (i * 256 + tid) * 4;  // 16 bytes per thread per iter
    const unsigned int lds_off =
        (unsigned int)(size_t)(const void*)&Xs[elt];
    const unsigned long long gaddr = (unsigned long long)(const void*)(Xb + elt);
    asm volatile("global_load_async_to_lds_b128 %0, %1, off"
                 :: "v"(lds_off), "v"(gaddr) : "memory");
  }
  asm volatile("s_wait_asynccnt 0x0" ::: "memory");
  __syncthreads();

  // ---- stage 1: Ys = M @ Xs   (wave owns a 16x128 row strip) ----
  {
    const int mrow = wave * 16;
    const int row = mrow + l16;
    v8f acc[8] = {};
    for (int k = 0; k < Dn; k += 4) {
      const int kk = k + (hi ? 2 : 0);
      v2f a;
      a.x = Mb[(size_t)row * Dn + kk];
      a.y = Mb[(size_t)row * Dn + kk + 1];
#pragma unroll
      for (int nt = 0; nt < 8; ++nt) {
        const int col = nt * 16 + l16;
        v2f bf;
        bf.x = Xs[kk * Dn + col];
        bf.y = Xs[(kk + 1) * Dn + col];
        acc[nt] = wmma4(a, bf, acc[nt]);
      }
    }
#pragma unroll
    for (int nt = 0; nt < 8; ++nt) {
      const int col = nt * 16 + l16;
      for (int r = 0; r < 8; ++r) {
        const int orow = mrow + r + (hi ? 8 : 0);
        Ys[orow * Dn + col] = acc[nt][r];
      }
    }
  }
  __syncthreads();

  // ---- stage 2: Xn = 2*Xs - Xs @ Ys   (A and B both from LDS) ----
  {
    const int mrow = wave * 16;
    const int row = mrow + l16;
    v8f acc[8] = {};
    for (int k = 0; k < Dn; k += 4) {
      const int kk = k + (hi ? 2 : 0);
      v2f a;
      a.x = Xs[row * Dn + kk];
      a.y = Xs[row * Dn + kk + 1];
#pragma unroll
      for (int nt = 0; nt < 8; ++nt) {
        const int col = nt * 16 + l16;
        v2f bf;
        bf.x = Ys[kk * Dn + col];
        bf.y = Ys[(kk + 1) * Dn + col];
        acc[nt] = wmma4(a, bf, acc[nt]);
      }
    }
#pragma unroll
    for (int nt = 0; nt < 8; ++nt) {
      const int col = nt * 16 + l16;
      for (int r = 0; r < 8; ++r) {
        const int orow = mrow + r + (hi ? 8 : 0);
        Xn[((size_t)b * Dn + orow) * Dn + col] =
            2.f * Xs[orow * Dn + col] - acc[nt][r];
      }
    }
  }
}

// prediction = M @ (M^+ @ tilde)
__global__ __launch_bounds__(Dn) void k_predict(
    const float* __restrict__ Mm, const float* __restrict__ Minv,
    const float* __restrict__ tilde, float* __restrict__ pred) {
  __shared__ float tmp[Dn];
  const int b = blockIdx.x, d = threadIdx.x;
  const float* tl = tilde + (size_t)b * Dn;
  float acc = 0.f;
  for (int k = 0; k < Dn; ++k) acc += Minv[((size_t)b * Dn + d) * Dn + k] * tl[k];
  tmp[d] = acc;
  __syncthreads();
  float p = 0.f;
  for (int k = 0; k < Dn; ++k) p += Mm[((size_t)b * Dn + d) * Dn + k] * tmp[k];
  pred[(size_t)b * Dn + d] = p;
}

// ===========================================================================
// Host launcher
// ===========================================================================
extern "C" void kernel_launch(void* const* d_in, const int* in_sizes, int n_in,
                              void* d_out, int out_size, void* d_ws, size_t ws_size,
                              hipStream_t stream) {
  const float* history  = (const float*)d_in[0];
  const int*   sidx     = (const int*)  d_in[1];
  const float* sig      = (const float*)d_in[2];
  const int*   cand     = (const int*)  d_in[3];
  const float* eps      = (const float*)d_in[4];
  const float* Wih      = (const float*)d_in[5];
  const float* Whh      = (const float*)d_in[6];
  const float* bih      = (const float*)d_in[7];
  const float* bhh      = (const float*)d_in[8];
  const float* Wmu      = (const float*)d_in[9];
  const float* bmu      = (const float*)d_in[10];
  const float* Wlv      = (const float*)d_in[11];
  const float* blv      = (const float*)d_in[12];
  const float* protos   = (const float*)d_in[13];
  const float* WcA      = (const float*)d_in[14];
  const float* WcB      = (const float*)d_in[15];
  const float* WcU      = (const float*)d_in[16];
  const float* Wpsi     = (const float*)d_in[17];
  const float* slin     = (const float*)d_in[18];
  const float* tlin     = (const float*)d_in[19];
  const float* sbias    = (const float*)d_in[20];
  const float* tbias    = (const float*)d_in[21];

  float* out = (float*)d_out;
  float* o_pred  = out;                       // B*D      = 2048
  float* o_trans = out + 2048;                // B*S*T    = 16384
  float* o_src   = out + 2048 + 16384;        // B*S      = 512
  float* o_tgt   = o_src + 512;               // B*T      = 512
  float* o_z     = o_tgt + 512;               // B*L      = 1024

  float* ws = (float*)d_ws;
  float* gi     = ws;                              // 16*128*384 = 786432
  float* summ   = gi + (size_t)Bn * STEPSn * 384;  // 2048
  float* zws    = summ + Bn * Dn;                  // 1024
  float* psiz   = zws + Bn * Ln;                   // 2048
  float* uz     = psiz + Bn * Dn;                  // 2048
  float* As     = uz + Bn * Dn;                    // 65536
  float* Bp     = As + Bn * Sn * Dn;               // 65536
  float* sa     = Bp + Bn * Tn * Dn;               // 512
  float* sb     = sa + Bn * Sn;                    // 512
  float* cost   = sb + Bn * Tn;                    // 16384
  float* inter  = cost + Bn * Sn * Tn;             // 65536
  float* basis  = inter + Bn * Sn * Dn;            // 2097152
  float* tilde  = basis + (size_t)Bn * Sn * Tn * Dn;  // 2048
  float* Mm     = tilde + Bn * Dn;                 // 262144
  float* X0     = Mm + (size_t)Bn * Dn * Dn;       // 262144
  float* X1     = X0 + (size_t)Bn * Dn * Dn;       // 262144

  // --- encoder ---
  k_gemm_gi<<<dim3((Bn * STEPSn) / 16, 384 / 16), 32, 0, stream>>>(history, Wih, bih, gi);
  k_gru_scan<<<Bn, Dn, 0, stream>>>(gi, Whh, bhh, summ);
  k_latent<<<Bn, Ln, 0, stream>>>(summ, Wmu, bmu, Wlv, blv, eps, zws, o_z);
  k_zproj<<<Bn, Dn, 0, stream>>>(zws, Wpsi, WcU, psiz, uz);

  // --- distributions ---
  k_src_dist<<<Bn, Sn, 0, stream>>>(sig, o_src);
  k_tgt_dist<<<Bn, Tn, 0, stream>>>(cand, protos, psiz, o_tgt);

  // --- cost matrix + Sinkhorn ---
  k_gemm_As<<<dim3((Bn * Sn) / 16, Dn / 16), 32, 0, stream>>>(sig, WcA, As);
  k_gemm_Bp<<<dim3((Bn * Tn) / 16, Dn / 16), 32, 0, stream>>>(cand, protos, WcB, uz, Bp);
  k_norms<<<Bn, 64, 0, stream>>>(As, Bp, sa, sb);
  k_gemm_cost<<<dim3(Bn, Sn / 16, Tn / 16), 32, 0, stream>>>(As, Bp, sa, sb, cost);
  k_sinkhorn<<<Bn, Sn * Tn, 0, stream>>>(cost, o_src, o_tgt, o_trans);

  // --- per-edge transforms + weighted sum ---
  k_inter<<<Bn * Sn, Dn, 0, stream>>>(sidx, sig, slin, inter);
  k_gemm_edge<<<dim3(Bn * Tn, Sn / 16), 32, 0, stream>>>(cand, inter, tlin, basis);
  k_tilde<<<Bn, Dn, 0, stream>>>(sidx, cand, sbias, tbias, o_trans, basis, tilde);

  // --- projection: M = basis^T basis; M^+ via Newton-Schulz; pred = M M^+ tilde
  k_gemm_M<<<dim3(Bn, Dn / 16, Dn / 16), 32, 0, stream>>>(basis, Mm);
  k_ns_init<<<Bn, Dn, 0, stream>>>(Mm, X0);
  float* Xcur = X0;
  float* Xnxt = X1;
  const size_t ns_lds = (size_t)2 * Dn * Dn * sizeof(float);  // 128 KB
  for (int it = 0; it < NS_ITERS; ++it) {
    k_ns_iter<<<Bn, 256, ns_lds, stream>>>(Mm, Xcur, Xnxt);
    float* tmp = Xcur; Xcur = Xnxt; Xnxt = tmp;
  }
  k_predict<<<Bn, Dn, 0, stream>>>(Mm, Xcur, tilde, o_pred);

  (void)in_sizes; (void)n_in; (void)out_size; (void)ws_size;
}